// Exp_loss_37168646980392
// MI455X (gfx1250) — compile-verified
//
#include <hip/hip_runtime.h>

typedef __attribute__((ext_vector_type(2))) float v2f;
typedef __attribute__((ext_vector_type(8))) float v8f;

#define S_ITEMS 128

// One wave (32 lanes) per assortment row; each lane handles 4 items.
// Produces lj[b] = log(exp(-s + floor(s)) / (S - rank)) - floor(s)
// (kept in the reference's exp/log form for numerical parity).
__global__ __launch_bounds__(256) void row_loss_kernel(
    const float* __restrict__ x, const float* __restrict__ y,
    const int* __restrict__ assort, float* __restrict__ lj, int B)
{
    const int wid  = threadIdx.x >> 5;
    const int lane = threadIdx.x & 31;
    const int b    = blockIdx.x * 8 + wid;
    if (b >= B) return;  // uniform per wave

    const int4 idx = *reinterpret_cast<const int4*>(assort + b * S_ITEMS + lane * 4);
    const float x0 = x[idx.x], x1 = x[idx.y], x2 = x[idx.z], x3 = x[idx.w];
    const float y0 = y[idx.x], y1 = y[idx.y], y2 = y[idx.z], y3 = y[idx.w];

    // chosen score = sum(x*y) over the row (y is one-hot within the row)
    float ch = x0 * y0 + x1 * y1 + x2 * y2 + x3 * y3;
    for (int off = 16; off; off >>= 1) ch += __shfl_xor(ch, off, 32);

    // s = sum relu(x - chosen); rank = #{x < chosen} (ascending argsort position)
    float s = fmaxf(x0 - ch, 0.0f) + fmaxf(x1 - ch, 0.0f) +
              fmaxf(x2 - ch, 0.0f) + fmaxf(x3 - ch, 0.0f);
    int cnt = (x0 < ch) + (x1 < ch) + (x2 < ch) + (x3 < ch);
    for (int off = 16; off; off >>= 1) {
        s   += __shfl_xor(s,   off, 32);
        cnt += __shfl_xor(cnt, off, 32);
    }

    if (lane == 0) {
        const float c = floorf(s);
        const float e = expf(-s + c);                 // argument in (-1, 0]
        const float t = e / (float)(S_ITEMS - cnt);   // S - rank >= 1 always
        lj[b] = logf(t) - c;
    }
}

// Single-wave final reduction of B floats via V_WMMA_F32_16X16X4_F32 with a
// ones B-matrix (matmul-as-rowsum). A-matrix 16x4 f32 lane layout per ISA:
// lanes 0-15 hold (M=lane, K=0..1), lanes 16-31 hold (M=lane-16, K=2..3).
__global__ __launch_bounds__(32) void final_reduce_wmma(
    const float* __restrict__ v, float* __restrict__ out, int nchunks, float invB)
{
    const int lane = threadIdx.x;
    const int col  = lane & 15;
    const int hi   = lane >> 4;

    v8f acc = {};
    v2f ones = {1.0f, 1.0f};

    for (int t = 0; t < nchunks; ++t) {
        const int base = t * 64 + col * 4 + hi * 2;   // A[m,k] = v[64t + 4m + k]
        v2f a = { v[base], v[base + 1] };
        // (neg_a, A, neg_b, B, c_mod, C, reuse_a, reuse_b)
        acc = __builtin_amdgcn_wmma_f32_16x16x4_f32(
            false, a, false, ones, (short)0, acc, false, false);
    }

    // D[m,n] identical across n; lanes 0-15 hold M=0..7 in VGPRs 0..7,
    // lanes 16-31 hold M=8..15. Fold VGPRs, then fold the two half-waves.
    float p = acc[0] + acc[1] + acc[2] + acc[3] + acc[4] + acc[5] + acc[6] + acc[7];
    p += __shfl_xor(p, 16, 32);
    if (lane == 0) out[0] = -p * invB;   // invB = 2^-11 -> exact, matches /B
}

extern "C" void kernel_launch(void* const* d_in, const int* in_sizes, int n_in,
                              void* d_out, int out_size, void* d_ws, size_t ws_size,
                              hipStream_t stream)
{
    const float* x      = (const float*)d_in[0];
    const float* y      = (const float*)d_in[1];
    const int*   assort = (const int*)d_in[2];

    const int B = in_sizes[2] / S_ITEMS;     // 2048
    float* lj = (float*)d_ws;                // B floats of scratch

    const int blocks = (B + 7) / 8;          // 8 waves (rows) per 256-thread block
    row_loss_kernel<<<blocks, 256, 0, stream>>>(x, y, assort, lj, B);

    const int nchunks = B / 64;              // 64 values consumed per WMMA chunk
    final_reduce_wmma<<<1, 32, 0, stream>>>(lj, (float*)d_out, nchunks,
                                            1.0f / (float)B);
}